// HungarianMatcher_42305427865906
// MI455X (gfx1250) — compile-verified
//
#include <hip/hip_runtime.h>
#include <hip/hip_bf16.h>

typedef __attribute__((ext_vector_type(2))) float v2f;
typedef __attribute__((ext_vector_type(8))) float v8f;

#define W_CLASS 2.0f
#define W_BBOX  5.0f
#define W_GIOU  2.0f

constexpr int NB = 32;     // batch
constexpr int NQ = 900;    // queries
constexpr int NC = 91;     // classes
constexpr int NG = 40;     // targets per image
constexpr int QPB = 128;   // q-rows per block (8 waves * 16)
constexpr int NT = 3;      // n-tiles of 16 -> 48 >= NG
constexpr int KSTEPS = 23; // 23*4 = 92 >= NC

__device__ __forceinline__ float fast_rcp(float x) {
    return __builtin_amdgcn_rcpf(x);   // v_rcp_f32, ~1 ulp
}
__device__ __forceinline__ float fast_sigmoid(float x) {
    return fast_rcp(1.0f + __expf(-x)); // v_exp_f32 + v_rcp_f32
}

__global__ __launch_bounds__(256)
void matcher_cost_kernel(const float* __restrict__ logits,   // (B,Q,C)
                         const float* __restrict__ pboxes,   // (B,Q,4) cxcywh
                         const int*   __restrict__ labels,   // (B,G)
                         const float* __restrict__ tboxes,   // (B,G,4) cxcywh
                         float* __restrict__ out)            // (B,Q,G)
{
    __shared__ int   s_lab[NT * 16];              // padded with -1
    __shared__ __align__(16) float s_tb[NG][4];   // raw cxcywh target boxes
    __shared__ __align__(16) float s_pb[QPB][4];  // raw cxcywh pred boxes

    const int b     = blockIdx.y;
    const int qbase = blockIdx.x * QPB;
    const int t     = threadIdx.x;

    // ---- stage per-batch target data + this block's pred boxes into LDS ----
    if (t < NT * 16) s_lab[t] = (t < NG) ? labels[b * NG + t] : -1;
    if (t >= 64 && t < 64 + NG) {
        const int g = t - 64;
        *(float4*)s_tb[g] = *(const float4*)(tboxes + ((size_t)b * NG + g) * 4);
    }
    if (t < QPB) {
        int q = qbase + t; if (q > NQ - 1) q = NQ - 1;
        *(float4*)s_pb[t] = *(const float4*)(pboxes + ((size_t)b * NQ + q) * 4);
    }
    __syncthreads();

    const int wave = t >> 5;
    const int lane = t & 31;
    const int q0   = qbase + wave * 16;     // this wave's 16-row q tile
    if (q0 >= NQ) return;                   // wave-uniform exit

    const int m  = lane & 15;
    const int hi = lane >> 4;

    // B-fragment columns: lane's n = (lane&15); preload the 3 labels it compares
    int labn[NT];
#pragma unroll
    for (int j = 0; j < NT; ++j) labn[j] = s_lab[j * 16 + m];

    // A-fragment row for this lane (clamped; padded rows never stored)
    int qa = q0 + m; if (qa > NQ - 1) qa = NQ - 1;
    const float* Arow = logits + ((size_t)b * NQ + qa) * NC;

    // ---- gather-as-GEMM: acc[j] += sigmoid(S) x onehot(E), f32 WMMA 16x16x4 ----
    // Fully unrolled: constant offsets, one-hot compares vs. inline constants.
    v8f acc[NT] = {};
#pragma unroll
    for (int ks = 0; ks < KSTEPS; ++ks) {
        const int k0 = ks * 4 + hi * 2;     // absolute class of a.x
        const int k1 = k0 + 1;              // absolute class of a.y
        const int k0c = (k0 > NC - 1) ? NC - 1 : k0;  // clamp OOB (finite * 0 = 0)
        const int k1c = (k1 > NC - 1) ? NC - 1 : k1;
        v2f a;
        a.x = fast_sigmoid(Arow[k0c]);
        a.y = fast_sigmoid(Arow[k1c]);
#pragma unroll
        for (int j = 0; j < NT; ++j) {
            v2f e;   // one-hot E[k,n]; labels < 91 so padded k==91 never matches
            e.x = (labn[j] == k0) ? 1.0f : 0.0f;
            e.y = (labn[j] == k1) ? 1.0f : 0.0f;
            acc[j] = __builtin_amdgcn_wmma_f32_16x16x4_f32(
                false, a, false, e, (short)0, acc[j], false, false);
        }
    }

    // ---- precompute per-lane target-box terms for its 3 columns ----
    float tcx[NT], tcy[NT], tw[NT], th[NT];
    float tx0[NT], ty0[NT], tx1[NT], ty1[NT], tarea[NT];
#pragma unroll
    for (int j = 0; j < NT; ++j) {
        const int g = j * 16 + m;
        const float4 tb = (g < NG) ? *(const float4*)s_tb[g]
                                   : make_float4(0.f, 0.f, 1.f, 1.f); // dummy, never stored
        tcx[j] = tb.x; tcy[j] = tb.y; tw[j] = tb.z; th[j] = tb.w;
        tx0[j] = tb.x - 0.5f * tb.z; ty0[j] = tb.y - 0.5f * tb.w;
        tx1[j] = tb.x + 0.5f * tb.z; ty1[j] = tb.y + 0.5f * tb.w;
        tarea[j] = tb.z * tb.w;
    }

    // ---- fuse L1 + GIoU per output element and store the final cost ----
    const int qr = wave * 16;  // row offset of this wave's tile inside s_pb
#pragma unroll
    for (int r = 0; r < 8; ++r) {
        const int M = r + hi * 8;           // C/D layout: lanes16-31 hold rows 8..15
        const int q = q0 + M;
        if (q >= NQ) continue;
        const float4 pb = *(const float4*)s_pb[qr + M];  // one LDS b128 per row
        const float pcx = pb.x, pcy = pb.y, pw = pb.z, ph = pb.w;
        const float px0 = pcx - 0.5f * pw, py0 = pcy - 0.5f * ph;
        const float px1 = pcx + 0.5f * pw, py1 = pcy + 0.5f * ph;
        const float parea = pw * ph;
#pragma unroll
        for (int j = 0; j < NT; ++j) {
            const int g = j * 16 + m;
            if (j == NT - 1 && g >= NG) continue;
            // L1 in cxcywh space
            const float l1 = fabsf(pcx - tcx[j]) + fabsf(pcy - tcy[j])
                           + fabsf(pw  - tw[j])  + fabsf(ph  - th[j]);
            // GIoU in xyxy space (fast rcp instead of IEEE divide)
            const float iw = fmaxf(fminf(px1, tx1[j]) - fmaxf(px0, tx0[j]), 0.0f);
            const float ih = fmaxf(fminf(py1, ty1[j]) - fmaxf(py0, ty0[j]), 0.0f);
            const float inter = iw * ih;
            const float uni   = parea + tarea[j] - inter;
            const float iou   = inter * fast_rcp(uni);
            const float cw = fmaxf(px1, tx1[j]) - fminf(px0, tx0[j]);
            const float ch = fmaxf(py1, ty1[j]) - fminf(py0, ty0[j]);
            const float carea = cw * ch;
            const float giou  = iou - (carea - uni) * fast_rcp(carea);
            const float cost  = W_BBOX * l1 - W_GIOU * giou - W_CLASS * acc[j][r];
            out[((size_t)b * NQ + q) * NG + g] = cost;
        }
    }
}

extern "C" void kernel_launch(void* const* d_in, const int* in_sizes, int n_in,
                              void* d_out, int out_size, void* d_ws, size_t ws_size,
                              hipStream_t stream) {
    (void)in_sizes; (void)n_in; (void)out_size; (void)d_ws; (void)ws_size;
    const float* logits = (const float*)d_in[0];   // pred_logits (B,Q,C) f32
    const float* pboxes = (const float*)d_in[1];   // pred_boxes  (B,Q,4) f32
    const int*   labels = (const int*)  d_in[2];   // tgt_labels  (B,G)   i32
    const float* tboxes = (const float*)d_in[3];   // tgt_boxes   (B,G,4) f32
    float* out = (float*)d_out;                    // (B,Q,G) f32

    dim3 block(256);
    dim3 grid((NQ + QPB - 1) / QPB, NB);           // (8, 32)
    matcher_cost_kernel<<<grid, block, 0, stream>>>(logits, pboxes, labels, tboxes, out);
}